// HeterogeneousGNN_90202903151245
// MI455X (gfx1250) — compile-verified
//
#include <hip/hip_runtime.h>

#define N_PAPER  50000
#define N_AUTHOR 20000
#define HDIM     128
#define IN_P     256
#define E_CITES  600000
#define E_WRITES 300000
#define BN_EPS   1e-5f

typedef __attribute__((ext_vector_type(2))) float v2f;
typedef __attribute__((ext_vector_type(8))) float v8f;

// ---------------------------------------------------------------------------
// C[M,128] = A[M,K] @ W[K,128] (+bias) (+relu), f32 WMMA 16x16x4.
// Block: 256 threads = 8 waves. blockIdx.x -> 16-row slab, wave -> 16-col tile.
// M % 16 == 0, K % 4 == 0 guaranteed by problem sizes.
// ---------------------------------------------------------------------------
__global__ __launch_bounds__(256) void gemm128_wmma_f32(
    const float* __restrict__ A, const float* __restrict__ W,
    const float* __restrict__ bias, float* __restrict__ C,
    int K, int relu)
{
    const int N   = 128;
    int wave = threadIdx.x >> 5;
    int lane = threadIdx.x & 31;
    int half = lane >> 4;        // 0: lanes 0-15, 1: lanes 16-31
    int l16  = lane & 15;
    int m0 = blockIdx.x << 4;
    int n0 = wave << 4;

    // A frag: lane holds M = l16, elements K = k0 + 2*half + {0,1}
    const float* Ap = A + (size_t)(m0 + l16) * K + (half << 1);
    // B frag: lane holds N = n0 + l16, elements K = k0 + 2*half + {0,1}
    const float* Wp = W + (size_t)(half << 1) * N + (n0 + l16);

    v8f acc = {};
    for (int k0 = 0; k0 < K; k0 += 4) {
        v2f a, b;
        a.x = Ap[k0];
        a.y = Ap[k0 + 1];
        b.x = Wp[(size_t)k0 * N];
        b.y = Wp[(size_t)(k0 + 1) * N];
        acc = __builtin_amdgcn_wmma_f32_16x16x4_f32(
            false, a, false, b, (short)0, acc, false, false);
    }

    int col = n0 + l16;
    float bb = bias ? bias[col] : 0.0f;
    float* Crow = C + (size_t)(m0 + (half << 3)) * N + col;
#pragma unroll
    for (int v = 0; v < 8; ++v) {
        float r = acc[v] + bb;
        if (relu) r = fmaxf(r, 0.0f);
        Crow[(size_t)v * N] = r;
    }
}

// ---------------------------------------------------------------------------
// Utility kernels
// ---------------------------------------------------------------------------
__global__ void fill_zero(float* __restrict__ p, int n) {
    int t = blockIdx.x * blockDim.x + threadIdx.x;
    if (t < n) p[t] = 0.0f;
}

// out[m*128+h] = scale * (b1[h] + (b2 ? b2[h] : 0))
__global__ void init_rows(float* __restrict__ out, const float* __restrict__ b1,
                          const float* __restrict__ b2, float scale, int M) {
    int t = blockIdx.x * blockDim.x + threadIdx.x;
    if (t >= M * HDIM) return;
    int h = t & (HDIM - 1);
    float v = b1[h];
    if (b2) v += b2[h];
    out[t] = v * scale;
}

__global__ void count_deg(const int* __restrict__ idx, float* __restrict__ deg, int E) {
    int t = blockIdx.x * blockDim.x + threadIdx.x;
    if (t < E) atomicAdd(&deg[idx[t]], 1.0f);
}

__global__ void inv_sqrt_deg(float* __restrict__ deg, int n) {
    int t = blockIdx.x * blockDim.x + threadIdx.x;
    if (t < n) {
        float d = deg[t];
        deg[t] = (d > 0.0f) ? rsqrtf(d) : 0.0f;
    }
}

// out[dst[e]] += scale * inv_s[src[e]] * inv_d[dst[e]] * msg[src[e]]
// 32 threads per edge, 4 floats per thread (H=128).
__global__ void scatter_edges(const float* __restrict__ msg,
                              const int* __restrict__ src, const int* __restrict__ dst,
                              const float* __restrict__ inv_s, const float* __restrict__ inv_d,
                              float* __restrict__ out, int E, float scale)
{
    int tid = blockIdx.x * blockDim.x + threadIdx.x;
    int e  = tid >> 5;
    int c4 = (tid & 31) << 2;
    if (e >= E) return;
    int s = src[e], d = dst[e];
    float coeff = inv_s[s] * inv_d[d] * scale;
    const float4 m = *(const float4*)(msg + (size_t)s * HDIM + c4);
    float* o = out + (size_t)d * HDIM + c4;
    atomicAdd(o + 0, coeff * m.x);
    atomicAdd(o + 1, coeff * m.y);
    atomicAdd(o + 2, coeff * m.z);
    atomicAdd(o + 3, coeff * m.w);
}

// Per-column sum / sumsq partials. Block = 128 threads (one per column),
// each block covers rowsPerBlock rows, then one atomicAdd per column per block.
__global__ void bn_stats(const float* __restrict__ x, float* __restrict__ sum,
                         float* __restrict__ sq, int M, int rowsPerBlock)
{
    int h  = threadIdx.x;
    int r0 = blockIdx.x * rowsPerBlock;
    int r1 = r0 + rowsPerBlock;
    if (r1 > M) r1 = M;
    float s = 0.0f, q = 0.0f;
    for (int r = r0; r < r1; ++r) {
        float v = x[(size_t)r * HDIM + h];
        s += v;
        q += v * v;
    }
    atomicAdd(&sum[h], s);
    atomicAdd(&sq[h], q);
}

// hout = relu(g*(x-mu)*rsqrt(var+eps)+b)  [+ hout if residual]
__global__ void bn_apply(const float* __restrict__ x, const float* __restrict__ sum,
                         const float* __restrict__ sq, const float* __restrict__ g,
                         const float* __restrict__ b, float* __restrict__ hout,
                         int M, int residual)
{
    int t = blockIdx.x * blockDim.x + threadIdx.x;
    if (t >= M * HDIM) return;
    int h = t & (HDIM - 1);
    float inv_m = 1.0f / (float)M;
    float mu  = sum[h] * inv_m;
    float var = sq[h] * inv_m - mu * mu;
    float y = g[h] * (x[t] - mu) * rsqrtf(var + BN_EPS) + b[h];
    y = fmaxf(y, 0.0f);
    hout[t] = residual ? (hout[t] + y) : y;
}

// ---------------------------------------------------------------------------
static inline int cdiv(int a, int b) { return (a + b - 1) / b; }

extern "C" void kernel_launch(void* const* d_in, const int* in_sizes, int n_in,
                              void* d_out, int out_size, void* d_ws, size_t ws_size,
                              hipStream_t stream)
{
    (void)in_sizes; (void)n_in; (void)out_size; (void)ws_size;

    const float* x_paper   = (const float*)d_in[0];
    const float* x_author  = (const float*)d_in[1];
    const int*   cites_src = (const int*)d_in[2];
    const int*   cites_dst = (const int*)d_in[3];
    const int*   writes_src= (const int*)d_in[4];
    const int*   writes_dst= (const int*)d_in[5];
    const int*   rev_src   = (const int*)d_in[6];
    const int*   rev_dst   = (const int*)d_in[7];
    // params flattened in dict insertion order starting at index 8
    const float* in_paper_W  = (const float*)d_in[8];
    const float* in_paper_b  = (const float*)d_in[9];
    const float* in_author_W = (const float*)d_in[10];
    const float* in_author_b = (const float*)d_in[11];
    const float* out_paper_W  = (const float*)d_in[32];
    const float* out_paper_b  = (const float*)d_in[33];
    const float* out_author_W = (const float*)d_in[34];
    const float* out_author_b = (const float*)d_in[35];

    // ---- workspace layout (floats) ----
    float* ws     = (float*)d_ws;
    float* h_p    = ws;
    float* h_a    = h_p   + (size_t)N_PAPER  * HDIM;
    float* msg_p  = h_a   + (size_t)N_AUTHOR * HDIM;
    float* msg_a  = msg_p + (size_t)N_PAPER  * HDIM;
    float* agg_p  = msg_a + (size_t)N_AUTHOR * HDIM;
    float* agg_a  = agg_p + (size_t)N_PAPER  * HDIM;
    float* inv_cs = agg_a + (size_t)N_AUTHOR * HDIM;
    float* inv_cd = inv_cs + N_PAPER;
    float* inv_ws_= inv_cd + N_PAPER;
    float* inv_wd = inv_ws_ + N_AUTHOR;
    float* inv_rs = inv_wd + N_PAPER;
    float* inv_rd = inv_rs + N_PAPER;
    float* bnsum  = inv_rd + N_AUTHOR;
    float* bnsq   = bnsum + HDIM;

    const int DEG_TOTAL = 4 * N_PAPER + 2 * N_AUTHOR;  // 240000

    // ---- input projections ----
    gemm128_wmma_f32<<<N_PAPER / 16, 256, 0, stream>>>(
        x_paper, in_paper_W, in_paper_b, h_p, IN_P, 1);
    gemm128_wmma_f32<<<N_AUTHOR / 16, 256, 0, stream>>>(
        x_author, in_author_W, in_author_b, h_a, HDIM, 1);

    // ---- degree normalizers (fixed across layers) ----
    fill_zero<<<cdiv(DEG_TOTAL, 256), 256, 0, stream>>>(inv_cs, DEG_TOTAL);
    count_deg<<<cdiv(E_CITES, 256), 256, 0, stream>>>(cites_src,  inv_cs,  E_CITES);
    count_deg<<<cdiv(E_CITES, 256), 256, 0, stream>>>(cites_dst,  inv_cd,  E_CITES);
    count_deg<<<cdiv(E_WRITES, 256), 256, 0, stream>>>(writes_src, inv_ws_, E_WRITES);
    count_deg<<<cdiv(E_WRITES, 256), 256, 0, stream>>>(writes_dst, inv_wd,  E_WRITES);
    count_deg<<<cdiv(E_WRITES, 256), 256, 0, stream>>>(rev_src,    inv_rs,  E_WRITES);
    count_deg<<<cdiv(E_WRITES, 256), 256, 0, stream>>>(rev_dst,    inv_rd,  E_WRITES);
    inv_sqrt_deg<<<cdiv(DEG_TOTAL, 256), 256, 0, stream>>>(inv_cs, DEG_TOTAL);

    // ---- GNN layers ----
    for (int l = 0; l < 2; ++l) {
        int pb = 12 + l * 10;
        const float* Wc  = (const float*)d_in[pb + 0];
        const float* bc  = (const float*)d_in[pb + 1];
        const float* Ww  = (const float*)d_in[pb + 2];
        const float* bw  = (const float*)d_in[pb + 3];
        const float* Wr  = (const float*)d_in[pb + 4];
        const float* br  = (const float*)d_in[pb + 5];
        const float* g_p = (const float*)d_in[pb + 6];
        const float* b_p = (const float*)d_in[pb + 7];
        const float* g_a = (const float*)d_in[pb + 8];
        const float* b_a = (const float*)d_in[pb + 9];

        // new_p = 0.5*(m_cites + m_writes); fold 0.5 into bias init + coeff.
        init_rows<<<cdiv(N_PAPER * HDIM, 256), 256, 0, stream>>>(
            agg_p, bc, bw, 0.5f, N_PAPER);
        init_rows<<<cdiv(N_AUTHOR * HDIM, 256), 256, 0, stream>>>(
            agg_a, br, nullptr, 1.0f, N_AUTHOR);

        // cites: paper -> paper
        gemm128_wmma_f32<<<N_PAPER / 16, 256, 0, stream>>>(
            h_p, Wc, nullptr, msg_p, HDIM, 0);
        scatter_edges<<<cdiv(E_CITES * 32, 256), 256, 0, stream>>>(
            msg_p, cites_src, cites_dst, inv_cs, inv_cd, agg_p, E_CITES, 0.5f);

        // writes: author -> paper
        gemm128_wmma_f32<<<N_AUTHOR / 16, 256, 0, stream>>>(
            h_a, Ww, nullptr, msg_a, HDIM, 0);
        scatter_edges<<<cdiv(E_WRITES * 32, 256), 256, 0, stream>>>(
            msg_a, writes_src, writes_dst, inv_ws_, inv_wd, agg_p, E_WRITES, 0.5f);

        // rev: paper -> author (h_p unchanged; reuse msg_p)
        gemm128_wmma_f32<<<N_PAPER / 16, 256, 0, stream>>>(
            h_p, Wr, nullptr, msg_p, HDIM, 0);
        scatter_edges<<<cdiv(E_WRITES * 32, 256), 256, 0, stream>>>(
            msg_p, rev_src, rev_dst, inv_rs, inv_rd, agg_a, E_WRITES, 1.0f);

        // BN + ReLU (+ residual for l>0) for paper
        fill_zero<<<1, 256, 0, stream>>>(bnsum, 2 * HDIM);
        bn_stats<<<cdiv(N_PAPER, 128), 128, 0, stream>>>(agg_p, bnsum, bnsq, N_PAPER, 128);
        bn_apply<<<cdiv(N_PAPER * HDIM, 256), 256, 0, stream>>>(
            agg_p, bnsum, bnsq, g_p, b_p, h_p, N_PAPER, l > 0 ? 1 : 0);

        // BN + ReLU (+ residual) for author
        fill_zero<<<1, 256, 0, stream>>>(bnsum, 2 * HDIM);
        bn_stats<<<cdiv(N_AUTHOR, 128), 128, 0, stream>>>(agg_a, bnsum, bnsq, N_AUTHOR, 128);
        bn_apply<<<cdiv(N_AUTHOR * HDIM, 256), 256, 0, stream>>>(
            agg_a, bnsum, bnsq, g_a, b_a, h_a, N_AUTHOR, l > 0 ? 1 : 0);
    }

    // ---- output projections (concatenated: paper then author) ----
    float* out_p = (float*)d_out;
    float* out_a = out_p + (size_t)N_PAPER * HDIM;
    gemm128_wmma_f32<<<N_PAPER / 16, 256, 0, stream>>>(
        h_p, out_paper_W, out_paper_b, out_p, HDIM, 0);
    gemm128_wmma_f32<<<N_AUTHOR / 16, 256, 0, stream>>>(
        h_a, out_author_W, out_author_b, out_a, HDIM, 0);
}